// QuantumLayer_42915313221864
// MI455X (gfx1250) — compile-verified
//
#include <hip/hip_runtime.h>
#include <math.h>

typedef __attribute__((ext_vector_type(2))) float v2f;
typedef __attribute__((ext_vector_type(8))) float v8f;

#define NQ 10
#define QD 4
#define PITCH 68   // LDS row pitch in floats: 68 % 64 == 4 -> conflict-free stride-PITCH access

// Workspace layout:
//   floats [0, 32768)  : 8 swizzled 64x64 operators (layer l, half h at (l*2+h)*4096)
//                        swizzle: pair-interleaved rows, float ofs = (row>>1)*128 + col*2 + (row&1)
//   bytes  [131072, +16K): int32 perm tables, 4 layers x 1024 src indices
#define WS_W_FLOATS 32768

// ---------------------------------------------------------------------------
// Kernel 0: build per-layer packed-real 64x64 operators (pre-swizzled for b64
// B-operand loads) and the per-layer composed CNOT-ring permutation tables.
//   U = Rot(w0) (x) ... (x) Rot(w4)  (32x32 complex tensor product)
//   W = [[Re(U^T), Im(U^T)], [-Im(U^T), Re(U^T)]]  (64x64 real, right-multiplier)
// ---------------------------------------------------------------------------
__device__ __forceinline__ void rot_entry(const float* qw, int l, int w,
                                          int i, int j, float& re, float& im) {
  // PennyLane Rot(phi, theta, omega) = RZ(omega) RY(theta) RZ(phi)
  float phi = qw[(l * NQ + w) * 3 + 0];
  float th  = qw[(l * NQ + w) * 3 + 1];
  float om  = qw[(l * NQ + w) * 3 + 2];
  float c   = cosf(0.5f * th);
  float s   = sinf(0.5f * th);
  float apo = 0.5f * (phi + om);
  float amo = 0.5f * (phi - om);
  if (i == 0 && j == 0)      { re =  cosf(apo) * c; im = -sinf(apo) * c; }
  else if (i == 0 && j == 1) { re = -cosf(amo) * s; im = -sinf(amo) * s; }
  else if (i == 1 && j == 0) { re =  cosf(amo) * s; im = -sinf(amo) * s; }
  else                       { re =  cosf(apo) * c; im =  sinf(apo) * c; }
}

__global__ void build_w_kernel(const float* __restrict__ qw, float* __restrict__ ws,
                               int* __restrict__ ptab) {
  int idx = blockIdx.x * 256 + threadIdx.x;
  if (idx < QD * 2 * 64 * 64) {
    // ---- operator matrices ----
    int l    = idx >> 13;
    int rem  = idx & 8191;
    int half = rem >> 12;
    int e    = rem & 4095;
    int row  = e >> 6;
    int col  = e & 63;
    int rr   = row & 31;
    int cc   = col & 31;
    // z = U[cc][rr]  (so W quadrant holds U^T[rr][cc])
    float zr = 1.f, zi = 0.f;
#pragma unroll
    for (int k = 0; k < 5; ++k) {
      int i = (cc >> (4 - k)) & 1;    // wire half*5+k <-> index bit (4-k)
      int j = (rr >> (4 - k)) & 1;
      float er, ei;
      rot_entry(qw, l, half * 5 + k, i, j, er, ei);
      float nr = zr * er - zi * ei;
      float ni = zr * ei + zi * er;
      zr = nr; zi = ni;
    }
    float val = (row < 32) ? ((col < 32) ? zr : zi)
                           : ((col < 32) ? -zi : zr);
    // pair-interleaved swizzle: rows 2p and 2p+1 interleave per column
    ws[(l * 2 + half) * 4096 + (row >> 1) * 128 + col * 2 + (row & 1)] = val;
  } else if (idx < QD * 2 * 64 * 64 + QD * 1024) {
    // ---- composed CNOT-ring permutation (gather source index) ----
    int e = idx - QD * 2 * 64 * 64;
    int l = e >> 10;
    int j = e & 1023;
    int r = (l % (NQ - 1)) + 1;
    int s = j;                          // src = f0(f1(...f9(j)))
#pragma unroll
    for (int w = NQ - 1; w >= 0; --w) {
      int cb = (NQ - 1) - w;            // control: wire w <-> bit 9-w
      int wr = w + r; if (wr >= NQ) wr -= NQ;
      int tb = (NQ - 1) - wr;           // target: wire (w+r)%10
      s ^= ((s >> cb) & 1) << tb;
    }
    ptab[e] = s;
  }
}

// ---------------------------------------------------------------------------
// (32x64) @ (64x64) real matmul on the f32 WMMA pipe, one wave per matmul.
//   src : LDS, rows M=0..31 at pitch PITCH, cols 0..63
//   W   : global, pair-interleaved swizzle (see build kernel), L2-resident
//   dst : LDS, stored TRANSPOSED within the complex packing:
//         dst[(c&31)][((c>=32)?32:0) + row] = D[row][c]
// ---------------------------------------------------------------------------
__device__ __forceinline__ void mm32x64(const float* src, const float* __restrict__ W,
                                        float* dst, int lane) {
  const int half = lane >> 4;  // 0: lanes 0-15, 1: lanes 16-31
  const int lm   = lane & 15;
  const v2f* __restrict__ Wp = (const v2f*)W;  // Wp[p*64 + n] = {W[2p][n], W[2p+1][n]}
  const v8f zero = {0.f, 0.f, 0.f, 0.f, 0.f, 0.f, 0.f, 0.f};
  v8f acc[2][4];
#pragma unroll
  for (int mi = 0; mi < 2; ++mi)
#pragma unroll
    for (int ni = 0; ni < 4; ++ni)
      acc[mi][ni] = zero;

#pragma unroll
  for (int k = 0; k < 16; ++k) {
    const int kb = 4 * k + 2 * half;  // this half-wave's K rows: kb, kb+1
    // A tiles (16x4 f32): lane<16 -> K=kb,kb+1 of row lm; lane>=16 -> K=4k+2,4k+3
    v2f a0 = *(const v2f*)(src + lm * PITCH + kb);
    v2f a1 = *(const v2f*)(src + (16 + lm) * PITCH + kb);
    // B tiles (4x16): one b64 load per fragment from the swizzled operator
    const int p = 2 * k + half;       // row-pair index
    v2f bt[4];
#pragma unroll
    for (int ni = 0; ni < 4; ++ni)
      bt[ni] = Wp[p * 64 + ni * 16 + lm];
#pragma unroll
    for (int ni = 0; ni < 4; ++ni) {
      acc[0][ni] = __builtin_amdgcn_wmma_f32_16x16x4_f32(
          false, a0, false, bt[ni], (short)0, acc[0][ni], false, false);
      acc[1][ni] = __builtin_amdgcn_wmma_f32_16x16x4_f32(
          false, a1, false, bt[ni], (short)0, acc[1][ni], false, false);
    }
  }

  // D layout: VGPR v holds D[M = v + 8*half][N = lane&15]; store transposed.
#pragma unroll
  for (int mi = 0; mi < 2; ++mi)
#pragma unroll
    for (int ni = 0; ni < 4; ++ni)
#pragma unroll
      for (int v = 0; v < 8; ++v) {
        int row = mi * 16 + v + half * 8;
        int c   = ni * 16 + lm;
        dst[(c & 31) * PITCH + ((c >> 5) << 5) + row] = acc[mi][ni][v];
      }
}

// ---------------------------------------------------------------------------
// Main kernel: one wave simulates one batch element, state lives in LDS.
// ---------------------------------------------------------------------------
__global__ void __launch_bounds__(64) qsim_kernel(const float* __restrict__ x,
                                                  const float* __restrict__ ws,
                                                  const int* __restrict__ ptab,
                                                  float* __restrict__ out, int B) {
  __shared__ float lds[2 * 2 * 32 * PITCH];  // 2 waves x (ping,pong) x 32xPITCH
  const int lane = threadIdx.x & 31;
  const int wv   = threadIdx.x >> 5;
  const int b    = blockIdx.x * 2 + wv;
  const int bb   = (b < B) ? b : (B - 1);   // clamp reads; stores are guarded
  float* bufA = lds + wv * (2 * 32 * PITCH);
  float* bufB = bufA + 32 * PITCH;

  // ---- RY angle embedding: amp[h][l] = A_h * B_l (real product state) ----
  float pa = 1.f, pb = 1.f;
#pragma unroll
  for (int k = 0; k < 5; ++k) {
    float xh = x[bb * NQ + k];         // wires 0..4 -> h bit (4-k)
    float xl = x[bb * NQ + 5 + k];     // wires 5..9 -> l bit (4-k)
    int bit = (lane >> (4 - k)) & 1;
    pa *= bit ? sinf(0.5f * xh) : cosf(0.5f * xh);
    pb *= bit ? sinf(0.5f * xl) : cosf(0.5f * xl);
  }
  for (int h = 0; h < 32; ++h) {
    float ah = __shfl(pa, h, 32);
    bufA[h * PITCH + lane]      = ah * pb;  // Re
    bufA[h * PITCH + 32 + lane] = 0.f;      // Im
  }

  float* cur = bufA;
  float* tmp = bufB;
  for (int l = 0; l < QD; ++l) {
    const float* Whigh = ws + (l * 2 + 0) * 4096;
    const float* Wlow  = ws + (l * 2 + 1) * 4096;
    const int*   pt    = ptab + l * 1024;
    __syncthreads();
    mm32x64(cur, Wlow, tmp, lane);   // S' = S @ Ulow^T      : [h][l] -> [l][h]
    __syncthreads();
    mm32x64(tmp, Whigh, cur, lane);  // S''^T = S'^T @ Uh^T  : [l][h] -> [h][l]
    __syncthreads();
    // ---- CNOT ring: gather through precomputed permutation, cur -> tmp ----
    for (int it = 0; it < 32; ++it) {
      int j = it * 32 + lane;          // dest index: h=it, l=lane
      int s = pt[j];                   // precomputed source index
      int hs = s >> 5, lsd = s & 31;
      float re = cur[hs * PITCH + lsd];
      float im = cur[hs * PITCH + 32 + lsd];
      tmp[it * PITCH + lane]      = re;
      tmp[it * PITCH + 32 + lane] = im;
    }
    float* t = cur; cur = tmp; tmp = t;
  }
  __syncthreads();

  // ---- <Z_i> = sum p * (+1 / -1) ----
  float ev[5] = {0.f, 0.f, 0.f, 0.f, 0.f};
  float lanesum = 0.f;
  for (int h = 0; h < 32; ++h) {
    float re = cur[h * PITCH + lane];
    float im = cur[h * PITCH + 32 + lane];
    float p  = re * re + im * im;
    lanesum += p;
#pragma unroll
    for (int k = 0; k < 5; ++k)
      ev[k] += ((h >> (4 - k)) & 1) ? -p : p;
  }
  float evl[10];
#pragma unroll
  for (int k = 0; k < 5; ++k) evl[k] = ev[k];
#pragma unroll
  for (int k = 0; k < 5; ++k)
    evl[5 + k] = ((lane >> (4 - k)) & 1) ? -lanesum : lanesum;
#pragma unroll
  for (int k = 0; k < 10; ++k) {
    float v = evl[k];
#pragma unroll
    for (int off = 16; off >= 1; off >>= 1)
      v += __shfl_xor(v, off, 32);
    evl[k] = v;
  }
  if (b < B && lane == 0) {
#pragma unroll
    for (int k = 0; k < 10; ++k) out[b * NQ + k] = evl[k];
  }
}

// ---------------------------------------------------------------------------
extern "C" void kernel_launch(void* const* d_in, const int* in_sizes, int n_in,
                              void* d_out, int out_size, void* d_ws, size_t ws_size,
                              hipStream_t stream) {
  const float* x  = (const float*)d_in[0];   // (B, 10) f32
  const float* qw = (const float*)d_in[1];   // (4, 10, 3) f32
  float* out = (float*)d_out;                // (B, 10) f32
  float* ws  = (float*)d_ws;                 // operators (128 KB) + perm tables (16 KB)
  int*   pt  = (int*)(ws + WS_W_FLOATS);
  int B = in_sizes[0] / NQ;

  int total0  = QD * 2 * 64 * 64 + QD * 1024;          // 36864 threads
  int blocks0 = (total0 + 255) / 256;
  hipLaunchKernelGGL(build_w_kernel, dim3(blocks0), dim3(256), 0, stream, qw, ws, pt);
  int blocks = (B + 1) / 2;                  // 2 waves (states) per 64-thread block
  hipLaunchKernelGGL(qsim_kernel, dim3(blocks), dim3(64), 0, stream, x, ws, pt, out, B);
}